// CoPE_17265768530810
// MI455X (gfx1250) — compile-verified
//
#include <hip/hip_runtime.h>
#include <math.h>

// CoPE gating, MI455X (gfx1250), wave32.
// Memory-bound op (~8 FLOP/byte): stream q once, write out once, keep the
// 256KB normalized codebook hot. Both einsums use V_WMMA_F32_16X16X4_F32
// (N=16 matches the tile exactly; f32 WMMA keeps exact reference precision).

typedef float v2f __attribute__((ext_vector_type(2)));
typedef float v8f __attribute__((ext_vector_type(8)));

#define DIM   4096
#define NPOS  16
#define ROWS  16384      // B*T = 4*4096
#define KC    64         // K-chunk per LDS stage
#define LDQ   68         // padded LDS stride (floats) -> conflict-free frag reads
#define GSTR  18         // padded gate-tile stride (floats)

// ---------------- Kernel 0: L2-normalize codebook rows ----------------
__global__ __launch_bounds__(256)
void cope_normalize(const float* __restrict__ pos, float* __restrict__ posv) {
    __shared__ float red[256];
    const int row = blockIdx.x;
    const float* p = pos + row * DIM;
    float s = 0.f;
    for (int i = threadIdx.x; i < DIM; i += 256) { float v = p[i]; s += v * v; }
    red[threadIdx.x] = s;
    __syncthreads();
    for (int off = 128; off > 0; off >>= 1) {
        if (threadIdx.x < off) red[threadIdx.x] += red[threadIdx.x + off];
        __syncthreads();
    }
    const float nrm = fmaxf(sqrtf(red[0]), 1e-12f);
    const float r = 1.f / nrm;
    float* o = posv + row * DIM;
    for (int i = threadIdx.x; i < DIM; i += 256) o[i] = p[i] * r;
}

// ---------------- Kernel 1: fused logits -> softmax -> mix ----------------
// Block = 128 threads (4 waves). Each wave owns 16 tokens; block owns 64.
__global__ __launch_bounds__(128)
void cope_main(const float* __restrict__ q,
               const float* __restrict__ posv,
               float* __restrict__ out) {
    __shared__ float qs[64 * LDQ];          // 64 tokens x KC, padded
    __shared__ float gs[4 * 16 * GSTR];     // per-wave 16x16 gate tiles, padded

    const int tid  = threadIdx.x;
    const int wave = tid >> 5;
    const int lane = tid & 31;
    const int half = lane >> 4;             // 0: lanes 0-15, 1: lanes 16-31
    const int l16  = lane & 15;
    const int c    = half * 2;              // K sub-offset per ISA A/B layout

    const int blockRow0 = blockIdx.x * 64;
    const int waveRow0  = blockRow0 + wave * 16;

    // ---------------- Phase A: logits[t][n] via WMMA f32 16x16x4 ----------
    v8f acc = {};
    // Per-lane base for B fragments: row n = l16 of posv, column sub-offset c.
    const float* pbase = posv + l16 * DIM + c;

    for (int k0 = 0; k0 < DIM; k0 += KC) {
        // Stage q[blockRow0..+63][k0..k0+KC) into LDS, fully coalesced b128.
        // 64*KC = 4096 floats = 1024 float4 / 128 threads = 8 each.
#pragma unroll
        for (int i = 0; i < 8; ++i) {
            const int idx = tid + i * 128;       // 0..1023
            const int r   = idx >> 4;            // row in tile (KC/4 = 16 f4/row)
            const int s4  = idx & 15;
            const float4 v = *(const float4*)(q + (size_t)(blockRow0 + r) * DIM + k0 + s4 * 4);
            *(float4*)&qs[r * LDQ + s4 * 4] = v;
        }
        __syncthreads();
#pragma unroll
        for (int kk = 0; kk < KC / 4; ++kk) {
            // A (q tile 16x4): lane -> m=l16, K = kk*4 + {c, c+1}
            v2f a = *(const v2f*)&qs[(wave * 16 + l16) * LDQ + kk * 4 + c];
            // B (posv^T 4x16): lane -> n=l16, K = kk*4 + {c, c+1}
            v2f b = *(const v2f*)(pbase + k0 + kk * 4);
            acc = __builtin_amdgcn_wmma_f32_16x16x4_f32(
                false, a, false, b, (short)0, acc, false, false);
        }
        __syncthreads();
    }

    // ---------------- Phase B: softmax over n (16 lanes per half) ----------
    // VGPR v: lanes 0-15 hold token t=v (n=lane); lanes 16-31 hold t=v+8.
    const float scale = 0.015625f;  // 1/sqrt(4096)
    float g[8];
#pragma unroll
    for (int v = 0; v < 8; ++v) {
        float x = acc[v] * scale;
        float m = x;
#pragma unroll
        for (int off = 1; off < 16; off <<= 1)
            m = fmaxf(m, __shfl_xor(m, off, 32));
        const float e = __expf(x - m);
        float s = e;
#pragma unroll
        for (int off = 1; off < 16; off <<= 1)
            s += __shfl_xor(s, off, 32);
        g[v] = e / s;
    }

    // Stash gates (per-wave region; same-wave RAW handled by s_wait_dscnt).
#pragma unroll
    for (int v = 0; v < 8; ++v)
        gs[(wave * 16 + v + half * 8) * GSTR + l16] = g[v];

    // ---------------- Phase C: out[t][d] = gates(16x16) @ posv(16x4096) ----
    // A fragments (gates, K = pos index) loaded once from LDS.
    v2f ga[4];
#pragma unroll
    for (int kn = 0; kn < 4; ++kn)
        ga[kn] = *(const v2f*)&gs[(wave * 16 + l16) * GSTR + kn * 4 + c];

    for (int d0 = 0; d0 < DIM; d0 += 16) {
        v8f o = {};
#pragma unroll
        for (int kn = 0; kn < 4; ++kn) {
            // B (posv 4x16): rows kn*4 + {c, c+1}, cols d0 + l16
            v2f b;
            b.x = posv[(size_t)(kn * 4 + c)     * DIM + d0 + l16];
            b.y = posv[(size_t)(kn * 4 + c + 1) * DIM + d0 + l16];
            o = __builtin_amdgcn_wmma_f32_16x16x4_f32(
                false, ga[kn], false, b, (short)0, o, false, false);
        }
#pragma unroll
        for (int v = 0; v < 8; ++v)
            out[(size_t)(waveRow0 + v + half * 8) * DIM + d0 + l16] = o[v];
    }
}

// ---------------- Host entry ----------------
extern "C" void kernel_launch(void* const* d_in, const int* in_sizes, int n_in,
                              void* d_out, int out_size, void* d_ws, size_t ws_size,
                              hipStream_t stream) {
    const float* q       = (const float*)d_in[0];
    // d_in[1] = x : unused by the reference computation
    const float* pos_emb = (const float*)d_in[2];
    float* out  = (float*)d_out;
    float* posv = (float*)d_ws;          // 16*4096 f32 = 256 KB scratch

    cope_normalize<<<NPOS, 256, 0, stream>>>(pos_emb, posv);
    cope_main<<<ROWS / 64, 128, 0, stream>>>(q, posv, out);
}